// GlobalModel_6244882448875
// MI455X (gfx1250) — compile-verified
//
#include <hip/hip_runtime.h>
#include <hip/hip_bf16.h>

typedef __attribute__((ext_vector_type(2))) float v2f;
typedef __attribute__((ext_vector_type(4))) float v4f;
typedef __attribute__((ext_vector_type(8))) float v8f;

#define D_NODE   256
#define D_GLOBAL 128
#define D_IN     384   // D_GLOBAL + D_NODE
#define HIDDEN   512
#define TARGETS  256
#define BATCHES  1024
#define NEG_SLOPE 0.01f
#define CHUNK    512   // nodes per scatter block

// ---------------------------------------------------------------------------
// Kernel 1: build hin[1024,384] = concat(u, 0)  (scatter kernel adds agg part)
// ---------------------------------------------------------------------------
__global__ void init_hin_kernel(const float* __restrict__ u, float* __restrict__ hin) {
    int idx = blockIdx.x * blockDim.x + threadIdx.x;
    if (idx >= BATCHES * D_IN) return;
    int row = idx / D_IN;
    int col = idx - row * D_IN;
    hin[idx] = (col < D_GLOBAL) ? u[row * D_GLOBAL + col] : 0.0f;
}

// ---------------------------------------------------------------------------
// Kernel 2: segment_sum of x[n,256] by sorted batch[] into hin[:,128:384].
// 256 threads/block. Thread layout: c = tid%64 owns 4 consecutive columns
// (b128 loads -> 512 B per wave per instruction), r = tid/64 is a row phase
// handling rows i = r, r+4, r+8, ... of the block's CHUNK-row window.
// x is read once (512 MB stream) -> non-temporal loads keep the 192 MB L2
// free for the 1.5 MB accumulator + GEMM weights. Sorted batch -> register
// run-accumulation, f32 global atomics only at segment boundaries (atomics
// commute, so the 4 interleaved row phases stay correct).
// ---------------------------------------------------------------------------
__global__ void scatter_x_kernel(const float* __restrict__ x,
                                 const int*   __restrict__ batch,
                                 float* __restrict__ hin, int n_nodes) {
    __shared__ int sbatch[CHUNK];
    const int start = blockIdx.x * CHUNK;
    const int tid = threadIdx.x;
    const int col4 = (tid & 63) << 2;          // first of 4 owned columns
    const int r    = tid >> 6;                 // row phase 0..3
    const int cnt = min(CHUNK, n_nodes - start);

    for (int i = tid; i < cnt; i += 256) sbatch[i] = batch[start + i];
    __syncthreads();

    if (r < cnt) {
        float a0 = 0.f, a1 = 0.f, a2 = 0.f, a3 = 0.f;
        int cur = sbatch[r];
        for (int i = r; i < cnt; i += 4) {
            int seg = sbatch[i];
            if (seg != cur) {
                float* dst = &hin[cur * D_IN + D_GLOBAL + col4];
                atomicAdd(dst + 0, a0);
                atomicAdd(dst + 1, a1);
                atomicAdd(dst + 2, a2);
                atomicAdd(dst + 3, a3);
                a0 = a1 = a2 = a3 = 0.f;
                cur = seg;
            }
            v4f v = __builtin_nontemporal_load(
                (const v4f*)(x + (size_t)(start + i) * D_NODE + col4));
            a0 += v.x; a1 += v.y; a2 += v.z; a3 += v.w;
        }
        float* dst = &hin[cur * D_IN + D_GLOBAL + col4];
        atomicAdd(dst + 0, a0);
        atomicAdd(dst + 1, a1);
        atomicAdd(dst + 2, a2);
        atomicAdd(dst + 3, a3);
    }
}

// ---------------------------------------------------------------------------
// Kernel 3: f32 WMMA GEMM:  out[M,N] = act(A[M,K] @ W[N,K]^T + bias[N])
// One wave per 16x16 output tile, V_WMMA_F32_16X16X4_F32 over K/4 steps.
// A frag (16x4, MxK): lanes 0-15 -> K = k,k+1 ; lanes 16-31 -> K = k+2,k+3.
// B frag (4x16, KxN): mirrored (W rows are the N dimension, contiguous in K).
// D (16x16): VGPR r -> M = r (lanes 0-15) / 8+r (lanes 16-31), lane%16 -> N.
// Grids are exact tile multiples so EXEC is all-ones for every WMMA.
// ---------------------------------------------------------------------------
template <bool ACT>
__global__ void wmma_gemm_kernel(const float* __restrict__ A,
                                 const float* __restrict__ W,
                                 const float* __restrict__ bias,
                                 float* __restrict__ out,
                                 int M, int N, int K) {
    const int wave  = (blockIdx.x * blockDim.x + threadIdx.x) >> 5;
    const int lane  = threadIdx.x & 31;
    const int tilesN = N >> 4;
    const int tm = (wave / tilesN) << 4;
    const int tn = (wave % tilesN) << 4;

    const int l16   = lane & 15;
    const int khalf = (lane >> 4) << 1;        // 0 or 2

    const float* arow = A + (size_t)(tm + l16) * K + khalf;
    const float* brow = W + (size_t)(tn + l16) * K + khalf;

    v8f c = {};
    for (int k = 0; k < K; k += 4) {
        v2f a = *(const v2f*)(arow + k);
        v2f b = *(const v2f*)(brow + k);
        // (neg_a, A, neg_b, B, c_mod, C, reuse_a, reuse_b)
        c = __builtin_amdgcn_wmma_f32_16x16x4_f32(false, a, false, b,
                                                  (short)0, c, false, false);
    }

    const float bv = bias[tn + l16];
    const int mbase = tm + ((lane >> 4) << 3); // +0 or +8
#pragma unroll
    for (int r = 0; r < 8; ++r) {
        float v = c[r] + bv;
        if (ACT) v = (v > 0.0f) ? v : NEG_SLOPE * v;
        out[(size_t)(mbase + r) * N + tn + l16] = v;
    }
}

// ---------------------------------------------------------------------------
// Launch: init -> scatter -> GEMM1(+lrelu) -> GEMM2(+lrelu) -> GEMM3
// ---------------------------------------------------------------------------
extern "C" void kernel_launch(void* const* d_in, const int* in_sizes, int n_in,
                              void* d_out, int out_size, void* d_ws, size_t ws_size,
                              hipStream_t stream) {
    const float* x     = (const float*)d_in[0];
    // d_in[1] edge_index, d_in[2] edge_attr: unused by the reference math
    const float* u     = (const float*)d_in[3];
    const int*   batch = (const int*)  d_in[4];
    const float* W1    = (const float*)d_in[5];
    const float* b1    = (const float*)d_in[6];
    const float* W2    = (const float*)d_in[7];
    const float* b2    = (const float*)d_in[8];
    const float* W3    = (const float*)d_in[9];
    const float* b3    = (const float*)d_in[10];

    const int n_nodes = in_sizes[0] / D_NODE;

    float* ws  = (float*)d_ws;
    float* hin = ws;                                   // [1024, 384]
    float* h1  = hin + (size_t)BATCHES * D_IN;         // [1024, 512]
    float* h2  = h1  + (size_t)BATCHES * HIDDEN;       // [1024, 512]

    // 1) hin = [u | 0]
    init_hin_kernel<<<(BATCHES * D_IN + 255) / 256, 256, 0, stream>>>(u, hin);

    // 2) hin[:,128:] += segment_sum(x, batch)
    const int nchunks = (n_nodes + CHUNK - 1) / CHUNK;
    scatter_x_kernel<<<nchunks, 256, 0, stream>>>(x, batch, hin, n_nodes);

    // 3) MLP via f32 WMMA (8 waves per 256-thread block, 1 tile per wave)
    const int t1 = (BATCHES / 16) * (HIDDEN  / 16);    // 2048 tiles
    const int t2 = (BATCHES / 16) * (HIDDEN  / 16);    // 2048 tiles
    const int t3 = (BATCHES / 16) * (TARGETS / 16);    // 1024 tiles

    wmma_gemm_kernel<true ><<<t1 / 8, 256, 0, stream>>>(hin, W1, b1, h1,
                                                        BATCHES, HIDDEN, D_IN);
    wmma_gemm_kernel<true ><<<t2 / 8, 256, 0, stream>>>(h1, W2, b2, h2,
                                                        BATCHES, HIDDEN, HIDDEN);
    wmma_gemm_kernel<false><<<t3 / 8, 256, 0, stream>>>(h2, W3, b3, (float*)d_out,
                                                        BATCHES, TARGETS, HIDDEN);
}